// DCN_59150289601021
// MI455X (gfx1250) — compile-verified
//
#include <hip/hip_runtime.h>
#include <math.h>

// Problem constants (from reference): B=4, C=64, O=64, H=W=128, K=3
#define HWSZ   16384      // 128*128
#define IMG    128
#define CIN    64
#define COUT   64
#define K2     9
#define CK     576        // CIN * K2
#define PIX_PER_BLK 256   // 2 image rows per block
#define NENT   2304       // 256 pixels * 9 taps per block

typedef float v2f __attribute__((ext_vector_type(2)));
typedef float v8f __attribute__((ext_vector_type(8)));

// ---------------------------------------------------------------------------
// Kernel 1: 3x3 offset conv (64 -> 18 channels), NCHW, pad=1, stride=1.
// One thread per (b,h,w) pixel, 18 accumulators; w_off reads are wave-uniform
// (scalarized by the compiler), x reads are lane-coalesced along w.
// ---------------------------------------------------------------------------
__global__ __launch_bounds__(256) void dcn_offset_conv(
    const float* __restrict__ x, const float* __restrict__ w_off,
    const float* __restrict__ b_off, float* __restrict__ off) {
  int p  = blockIdx.x * blockDim.x + threadIdx.x;   // 0 .. 65535
  int b  = p >> 14;
  int hw = p & (HWSZ - 1);
  int h  = hw >> 7;
  int w  = hw & 127;

  float acc[18];
#pragma unroll
  for (int co = 0; co < 18; ++co) acc[co] = b_off[co];

  const float* xb = x + (size_t)b * CIN * HWSZ;
  for (int kh = 0; kh < 3; ++kh) {
    int yy = h + kh - 1;
    if (yy < 0 || yy >= IMG) continue;
    for (int kw = 0; kw < 3; ++kw) {
      int xx = w + kw - 1;
      if (xx < 0 || xx >= IMG) continue;
      const float* xp = xb + yy * IMG + xx;
      const float* wp = w_off + kh * 3 + kw;     // [co*576 + ci*9 + kh*3+kw]
      for (int ci = 0; ci < CIN; ++ci) {
        float xv = xp[(size_t)ci * HWSZ];
#pragma unroll
        for (int co = 0; co < 18; ++co)
          acc[co] = fmaf(xv, wp[co * CK + ci * K2], acc[co]);
      }
    }
  }
#pragma unroll
  for (int co = 0; co < 18; ++co)
    off[(((size_t)b * 18 + co) << 14) + hw] = acc[co];
}

// ---------------------------------------------------------------------------
// Kernel 2: fused bilinear-sample + WMMA GEMM + bias.
// Block = 256 threads (8 wave32), covers 2 image rows = 256 pixels.
// Phase 1: LDS table of 4 gather indices + 4 masked bilinear weights per
//          (pixel, tap) : 2304 entries x 32 B = 73.7 KB.
// Phase 2: each wave owns TWO 16-pixel N-tiles x all 64 output channels:
//          the 4 A fragments (weights) feed 8 V_WMMA_F32_16X16X4_F32 per
//          K-step (2.5 loads/WMMA instead of 3), K = 576 in steps of 4.
// ---------------------------------------------------------------------------
__global__ __launch_bounds__(256) void dcn_wmma_gemm(
    const float* __restrict__ x, const float* __restrict__ off,
    const float* __restrict__ wmat,   // (64, 576) = w flattened [o][c*9+k]
    const float* __restrict__ bias, float* __restrict__ out) {
  __shared__ __align__(16) int   sIdx[NENT * 4];
  __shared__ __align__(16) float sWgt[NENT * 4];

  const int pixbase = blockIdx.x * PIX_PER_BLK;   // 256 blocks
  const int b       = pixbase >> 14;              // batch constant per block

  // ---- Phase 1: tap table (256 pixels x 9 taps) ----
  for (int e = threadIdx.x; e < NENT; e += 256) {
    int pl = e / 9;                 // pixel local 0..255
    int k  = e - pl * 9;            // tap 0..8
    int kh = k / 3;
    int kw = k - kh * 3;
    int hw = (pixbase + pl) & (HWSZ - 1);
    int h  = hw >> 7;
    int w  = hw & 127;
    float oy = off[(((size_t)b * 18 + 2 * k)     << 14) + hw];
    float ox = off[(((size_t)b * 18 + 2 * k + 1) << 14) + hw];
    float y  = (float)(h + kh - 1) + oy;      // base_y + oy
    float xf = (float)(w + kw - 1) + ox;      // base_x + ox
    float y0f = floorf(y), x0f = floorf(xf);
    float wy = y - y0f, wx = xf - x0f;
    int y0 = (int)y0f, x0 = (int)x0f;
    int y1 = y0 + 1,  x1 = x0 + 1;
    float vy0 = (y0 >= 0 && y0 < IMG) ? 1.f : 0.f;
    float vy1 = (y1 >= 0 && y1 < IMG) ? 1.f : 0.f;
    float vx0 = (x0 >= 0 && x0 < IMG) ? 1.f : 0.f;
    float vx1 = (x1 >= 0 && x1 < IMG) ? 1.f : 0.f;
    int y0c = min(max(y0, 0), IMG - 1), y1c = min(max(y1, 0), IMG - 1);
    int x0c = min(max(x0, 0), IMG - 1), x1c = min(max(x1, 0), IMG - 1);
    int be = e * 4;
    sIdx[be + 0] = y0c * IMG + x0c;  sWgt[be + 0] = (1.f - wy) * (1.f - wx) * vy0 * vx0;
    sIdx[be + 1] = y0c * IMG + x1c;  sWgt[be + 1] = (1.f - wy) * wx * vy0 * vx1;
    sIdx[be + 2] = y1c * IMG + x0c;  sWgt[be + 2] = wy * (1.f - wx) * vy1 * vx0;
    sIdx[be + 3] = y1c * IMG + x1c;  sWgt[be + 3] = wy * wx * vy1 * vx1;
  }
  __syncthreads();

  // ---- Phase 2: WMMA GEMM, 2 N-tiles per wave ----
  const int lane = threadIdx.x & 31;
  const int wv   = threadIdx.x >> 5;   // wave id 0..7
  const int lo   = lane & 15;          // N column / M row within fragment
  const int hi   = lane >> 4;          // lane-half selects K pair / M half
  const int pl0  = wv * 32 + lo;       // pixel of N-tile 0
  const int pl1  = pl0 + 16;           // pixel of N-tile 1
  const float* xb = x + (size_t)b * CIN * HWSZ;

  v8f acc[8];                          // [tile(2)][mtile(4)]
  const v8f vzero = {0.f, 0.f, 0.f, 0.f, 0.f, 0.f, 0.f, 0.f};
#pragma unroll
  for (int i = 0; i < 8; ++i) acc[i] = vzero;

  for (int ks = 0; ks < CK / 4; ++ks) {
    const int kbase = ks * 4;
    const int ck0 = kbase + hi * 2;      // this lane-half's K rows
    const int ck1 = ck0 + 1;
    const int c0 = ck0 / 9, k0 = ck0 - c0 * 9;
    const int c1 = ck1 / 9, k1 = ck1 - c1 * 9;
    const float* xc0 = xb + (size_t)c0 * HWSZ;
    const float* xc1 = xb + (size_t)c1 * HWSZ;

    // B fragment, tile 0: S[ck0..ck1][pixel pl0]
    {
      const int    e0 = (pl0 * 9 + k0) * 4;
      const int4   i0 = *reinterpret_cast<const int4*>(&sIdx[e0]);
      const float4 f0 = *reinterpret_cast<const float4*>(&sWgt[e0]);
      const int    e1 = (pl0 * 9 + k1) * 4;
      const int4   i1 = *reinterpret_cast<const int4*>(&sIdx[e1]);
      const float4 f1 = *reinterpret_cast<const float4*>(&sWgt[e1]);
      v2f bf;
      bf.x = f0.x * xc0[i0.x] + f0.y * xc0[i0.y] + f0.z * xc0[i0.z] + f0.w * xc0[i0.w];
      bf.y = f1.x * xc1[i1.x] + f1.y * xc1[i1.y] + f1.z * xc1[i1.z] + f1.w * xc1[i1.w];
#pragma unroll
      for (int mt = 0; mt < 4; ++mt) {
        v2f a = *reinterpret_cast<const v2f*>(&wmat[(size_t)(mt * 16 + lo) * CK + ck0]);
        acc[mt] = __builtin_amdgcn_wmma_f32_16x16x4_f32(
            false, a, false, bf, (short)0, acc[mt], false, false);
      }
    }
    // B fragment, tile 1: S[ck0..ck1][pixel pl1]  (A fragments re-fetched from
    // the same addresses -> served by L0, amortizing weight traffic)
    {
      const int    e0 = (pl1 * 9 + k0) * 4;
      const int4   i0 = *reinterpret_cast<const int4*>(&sIdx[e0]);
      const float4 f0 = *reinterpret_cast<const float4*>(&sWgt[e0]);
      const int    e1 = (pl1 * 9 + k1) * 4;
      const int4   i1 = *reinterpret_cast<const int4*>(&sIdx[e1]);
      const float4 f1 = *reinterpret_cast<const float4*>(&sWgt[e1]);
      v2f bf;
      bf.x = f0.x * xc0[i0.x] + f0.y * xc0[i0.y] + f0.z * xc0[i0.z] + f0.w * xc0[i0.w];
      bf.y = f1.x * xc1[i1.x] + f1.y * xc1[i1.y] + f1.z * xc1[i1.z] + f1.w * xc1[i1.w];
#pragma unroll
      for (int mt = 0; mt < 4; ++mt) {
        v2f a = *reinterpret_cast<const v2f*>(&wmat[(size_t)(mt * 16 + lo) * CK + ck0]);
        acc[4 + mt] = __builtin_amdgcn_wmma_f32_16x16x4_f32(
            false, a, false, bf, (short)0, acc[4 + mt], false, false);
      }
    }
  }

  // ---- Epilogue: bias + store. C/D layout: VGPR r -> M = 16*mt + 8*hi + r, N = lo
#pragma unroll
  for (int t = 0; t < 2; ++t) {
    const int hw = (pixbase + (t == 0 ? pl0 : pl1)) & (HWSZ - 1);
#pragma unroll
    for (int mt = 0; mt < 4; ++mt) {
#pragma unroll
      for (int r = 0; r < 8; ++r) {
        int m = mt * 16 + hi * 8 + r;
        out[((size_t)b * COUT + m) * HWSZ + hw] = acc[t * 4 + mt][r] + bias[m];
      }
    }
  }
}

// ---------------------------------------------------------------------------
extern "C" void kernel_launch(void* const* d_in, const int* in_sizes, int n_in,
                              void* d_out, int out_size, void* d_ws, size_t ws_size,
                              hipStream_t stream) {
  (void)in_sizes; (void)n_in; (void)out_size; (void)ws_size;
  const float* x     = (const float*)d_in[0];
  const float* w_off = (const float*)d_in[1];
  const float* b_off = (const float*)d_in[2];
  const float* w     = (const float*)d_in[3];
  const float* bias  = (const float*)d_in[4];
  float* out = (float*)d_out;
  float* off = (float*)d_ws;   // (4, 18, 128, 128) fp32 = 4.72 MB scratch

  // 65536 pixels, one thread each
  dcn_offset_conv<<<256, 256, 0, stream>>>(x, w_off, b_off, off);
  // 256 blocks x 256 threads: two image rows per block, 8 waves x 2 N-tiles
  dcn_wmma_gemm<<<256, 256, 0, stream>>>(x, off, w, bias, out);
}